// MambaBlock_29351806500921
// MI455X (gfx1250) — compile-verified
//
#include <hip/hip_runtime.h>
#include <hip/hip_bf16.h>

typedef __attribute__((ext_vector_type(2))) float v2f;
typedef __attribute__((ext_vector_type(8))) float v8f;

#define MODEL_DIM 1024
#define D_STATE   16
#define D_CONV    4
#define D_INNER   2048
#define BATCH     4
#define SEQLEN    1024
#define NROWS     (BATCH * SEQLEN)   // 4096

__device__ __forceinline__ float silu_f(float x) {
    return x / (1.0f + __expf(-x));
}

// ---------------------------------------------------------------------------
// Kernel 1/6: C[M,N] = A[M,K] @ W[K,N], row-major fp32, V_WMMA_F32_16X16X4_F32.
// Block = 256 threads (8 waves stacked in M); block tile 256(M) x 128(N);
// wave tile 32(M) x 128(N), acc = 16 x v8f = 128 VGPRs.
// W chunk (16 x 128) staged through LDS in pair-interleaved layout
//   Bs[k2*288 + n*2 + (k&1)]  (k2 = k/2, pitch 288 dwords: 288 % 64 == 32, so
//   the khalf=0 and khalf=2 lane halves hit disjoint LDS bank halves)
// so each B fragment is a single aligned ds_load_b64 (no repack movs).
// Next chunk prefetched into registers during compute; single LDS buffer.
// Requires M%256==0, N%128==0, K%16==0.
// ---------------------------------------------------------------------------
#define KC 16
#define BPITCH 288                                // dwords per k-pair row
__global__ void __launch_bounds__(256)
wmma_gemm_f32(const float* __restrict__ A, const float* __restrict__ W,
              float* __restrict__ C, int M, int N, int K)
{
    __shared__ float Bs[(KC / 2) * BPITCH];       // 8*288*4 = 9216 B

    const int wave   = threadIdx.x >> 5;          // 0..7 -> M sub-tile
    const int lane   = threadIdx.x & 31;
    const int lane16 = lane & 15;
    const int khalf  = (lane >> 4) * 2;           // 0 or 2 (A/B K split)

    const int m0 = blockIdx.y * 256 + wave * 32;
    const int n0 = blockIdx.x * 128;

    const float* Arow0 = A + (size_t)(m0 + lane16) * K;        // rows m0..m0+15
    const float* Arow1 = A + (size_t)(m0 + 16 + lane16) * K;   // rows m0+16..m0+31

    // Cooperative W staging: wave w owns rows w and w+8 of the 16x128 chunk,
    // lane covers cols lane*4 .. lane*4+3 (coalesced 512B per row) -> LDS
    // pair-interleaved scatter.
    const int wr = wave;                          // 0..7
    const int wc = lane * 4;                      // 0..124
    const float* Wbase = W + (size_t)wr * N + n0 + wc;
    float* Bst0 = Bs + (wr >> 1) * BPITCH + (wr & 1);        // row wr
    float* Bst1 = Bs + ((wr + 8) >> 1) * BPITCH + (wr & 1);  // row wr+8
    float4 w0, w1;

    v8f acc[2][8];
    #pragma unroll
    for (int g = 0; g < 2; ++g)
        #pragma unroll
        for (int j = 0; j < 8; ++j)
            acc[g][j] = (v8f){};

    // preload chunk 0
    w0 = *(const float4*)(Wbase);
    w1 = *(const float4*)(Wbase + (size_t)8 * N);
    {
        Bst0[(wc + 0) * 2] = w0.x;  Bst0[(wc + 1) * 2] = w0.y;
        Bst0[(wc + 2) * 2] = w0.z;  Bst0[(wc + 3) * 2] = w0.w;
        Bst1[(wc + 0) * 2] = w1.x;  Bst1[(wc + 1) * 2] = w1.y;
        Bst1[(wc + 2) * 2] = w1.z;  Bst1[(wc + 3) * 2] = w1.w;
    }
    __syncthreads();

    for (int k = 0; k < K; k += KC) {
        const bool has_next = (k + KC) < K;
        if (has_next) {
            const float* Wp = Wbase + (size_t)(k + KC) * N;
            w0 = *(const float4*)(Wp);
            w1 = *(const float4*)(Wp + (size_t)8 * N);
        }

        #pragma unroll
        for (int kk = 0; kk < KC; kk += 4) {
            v2f a0 = *(const v2f*)(Arow0 + k + kk + khalf);
            v2f a1 = *(const v2f*)(Arow1 + k + kk + khalf);
            // fragment base: k-pair row (kk+khalf)/2, column lane16
            const float* bp = Bs + ((kk + khalf) >> 1) * BPITCH + lane16 * 2;
            #pragma unroll
            for (int j = 0; j < 8; ++j) {
                v2f b = *(const v2f*)(bp + j * 32);   // {W[k,n], W[k+1,n]}
                acc[0][j] = __builtin_amdgcn_wmma_f32_16x16x4_f32(
                    false, a0, false, b, (short)0, acc[0][j], false, false);
                acc[1][j] = __builtin_amdgcn_wmma_f32_16x16x4_f32(
                    false, a1, false, b, (short)0, acc[1][j], false, false);
            }
        }

        __syncthreads();
        if (has_next) {
            Bst0[(wc + 0) * 2] = w0.x;  Bst0[(wc + 1) * 2] = w0.y;
            Bst0[(wc + 2) * 2] = w0.z;  Bst0[(wc + 3) * 2] = w0.w;
            Bst1[(wc + 0) * 2] = w1.x;  Bst1[(wc + 1) * 2] = w1.y;
            Bst1[(wc + 2) * 2] = w1.z;  Bst1[(wc + 3) * 2] = w1.w;
            __syncthreads();
        }
    }

    // C/D layout: VGPR r -> row r (lanes 0-15) / row r+8 (lanes 16-31), col = lane&15
    const int crow = (lane >> 4) ? 8 : 0;
    const int ccol = n0 + lane16;
    #pragma unroll
    for (int g = 0; g < 2; ++g) {
        float* Cp = C + (size_t)(m0 + g * 16 + crow) * N + ccol;
        #pragma unroll
        for (int r = 0; r < 8; ++r) {
            #pragma unroll
            for (int j = 0; j < 8; ++j)
                Cp[(size_t)r * N + j * 16] = acc[g][j][r];
        }
    }
}

// ---------------------------------------------------------------------------
// Kernel 2/6: causal depthwise conv (D_CONV=4) + SiLU.
// x_main lives in xz[:, 0:2048]; output x_conv[row, d] (row = b*L + l).
// ---------------------------------------------------------------------------
__global__ void __launch_bounds__(256)
conv_silu_kernel(const float* __restrict__ xz,
                 const float* __restrict__ conv_w,
                 const float* __restrict__ conv_b,
                 float* __restrict__ xconv)
{
    size_t idx = (size_t)blockIdx.x * blockDim.x + threadIdx.x;  // row*2048 + d
    if (idx >= (size_t)NROWS * D_INNER) return;
    const int d = (int)(idx & (D_INNER - 1));
    const size_t row = idx >> 11;                 // b*L + l
    const int l = (int)(row & (SEQLEN - 1));

    float acc = conv_b[d];
    const float* w = conv_w + (size_t)d * D_CONV; // [D_INNER, D_CONV]
    #pragma unroll
    for (int k = 0; k < D_CONV; ++k) {
        int li = l - (D_CONV - 1) + k;
        if (li >= 0)
            acc += xz[(row - l + li) * (2 * D_INNER) + d] * w[k];
    }
    xconv[idx] = silu_f(acc);
}

// ---------------------------------------------------------------------------
// Kernel 3/6: ssm[row, 0:33] = x_conv[row, :] @ W_x  (W_x is [2048, 33]).
// One 256-thread block per row; row staged in LDS; wave-per-column dot with
// shuffle reduction.
// ---------------------------------------------------------------------------
__global__ void __launch_bounds__(256)
ssm_proj_kernel(const float* __restrict__ xconv,
                const float* __restrict__ Wx,
                float* __restrict__ ssm)
{
    __shared__ float xs[D_INNER];
    const int row = blockIdx.x;
    const float* xr = xconv + (size_t)row * D_INNER;
    for (int d = threadIdx.x; d < D_INNER; d += 256) xs[d] = xr[d];
    __syncthreads();

    const int wave = threadIdx.x >> 5;
    const int lane = threadIdx.x & 31;
    const int NC = 2 * D_STATE + 1;              // 33
    for (int j = wave; j < NC; j += 8) {
        float s = 0.0f;
        for (int d = lane; d < D_INNER; d += 32)
            s += xs[d] * Wx[(size_t)d * NC + j];
        #pragma unroll
        for (int off = 16; off > 0; off >>= 1)
            s += __shfl_down(s, off, 32);
        if (lane == 0) ssm[(size_t)row * NC + j] = s;
    }
}

// ---------------------------------------------------------------------------
// Kernel 4/6: selective scan. One thread per (b, d) channel; h[16] and A[16]
// in registers; ssm rows staged into LDS in 64-step chunks (33 floats/step).
// ---------------------------------------------------------------------------
#define TCH 64
__global__ void __launch_bounds__(64)
scan_kernel(const float* __restrict__ xconv,
            const float* __restrict__ ssm,
            const float* __restrict__ dt_w,
            const float* __restrict__ dt_b,
            const float* __restrict__ A_log,
            const float* __restrict__ Dp,
            float* __restrict__ yinner)
{
    __shared__ float sb[TCH * 33];
    const int d = blockIdx.x * 64 + threadIdx.x;   // 0..2047
    const int b = blockIdx.y;

    const float dw = dt_w[d];
    const float db = dt_b[d];
    const float Dd = Dp[d];

    float Areg[D_STATE];
    #pragma unroll
    for (int n = 0; n < D_STATE; ++n)
        Areg[n] = -__expf(A_log[(size_t)d * D_STATE + n]);

    float h[D_STATE];
    #pragma unroll
    for (int n = 0; n < D_STATE; ++n) h[n] = 0.0f;

    for (int t0 = 0; t0 < SEQLEN; t0 += TCH) {
        __syncthreads();
        {
            const float* src = ssm + (size_t)(b * SEQLEN + t0) * 33;
            for (int i = threadIdx.x; i < TCH * 33; i += 64) sb[i] = src[i];
        }
        __syncthreads();

        for (int tt = 0; tt < TCH; ++tt) {
            const float* r = sb + tt * 33;
            const size_t row = (size_t)(b * SEQLEN + t0 + tt);
            const float draw = r[0];
            const float xv = xconv[row * D_INNER + d];

            float zz = dw * draw + db;
            float delta = (zz > 20.0f) ? zz : log1pf(__expf(zz));   // softplus
            const float dx = delta * xv;

            float y = 0.0f;
            #pragma unroll
            for (int n = 0; n < D_STATE; ++n) {
                float dA = __expf(delta * Areg[n]);
                h[n] = dA * h[n] + dx * r[1 + n];
                y += h[n] * r[1 + D_STATE + n];
            }
            yinner[row * D_INNER + d] = y + Dd * xv;
        }
    }
}

// ---------------------------------------------------------------------------
// Kernel 5/6: y_gated = y_inner * silu(z),  z = xz[:, 2048:4096]
// ---------------------------------------------------------------------------
__global__ void __launch_bounds__(256)
gate_kernel(const float* __restrict__ yinner,
            const float* __restrict__ xz,
            float* __restrict__ ygated)
{
    size_t idx = (size_t)blockIdx.x * blockDim.x + threadIdx.x;  // row*2048 + d
    if (idx >= (size_t)NROWS * D_INNER) return;
    const int d = (int)(idx & (D_INNER - 1));
    const size_t row = idx >> 11;
    const float z = xz[row * (2 * D_INNER) + D_INNER + d];
    ygated[idx] = yinner[idx] * silu_f(z);
}

// ---------------------------------------------------------------------------
// Launch
// ---------------------------------------------------------------------------
extern "C" void kernel_launch(void* const* d_in, const int* in_sizes, int n_in,
                              void* d_out, int out_size, void* d_ws, size_t ws_size,
                              hipStream_t stream)
{
    const float* x      = (const float*)d_in[0];
    const float* W_in   = (const float*)d_in[1];
    const float* conv_w = (const float*)d_in[2];
    const float* conv_b = (const float*)d_in[3];
    const float* W_x    = (const float*)d_in[4];
    const float* dt_w   = (const float*)d_in[5];
    const float* dt_b   = (const float*)d_in[6];
    const float* A_log  = (const float*)d_in[7];
    const float* Dp     = (const float*)d_in[8];
    const float* W_out  = (const float*)d_in[9];
    float* out = (float*)d_out;

    float* ws = (float*)d_ws;
    float* xz     = ws;                                   // 4096*4096
    float* xconv  = xz    + (size_t)NROWS * 2 * D_INNER;  // 4096*2048
    float* ssm    = xconv + (size_t)NROWS * D_INNER;      // 4096*33
    float* yinner = ssm   + (size_t)NROWS * 33;           // 4096*2048
    float* ygated = xconv;                                // reuse (xconv dead after scan)

    // 1) xz = x @ W_in  (M=4096, N=4096, K=1024)
    wmma_gemm_f32<<<dim3(2 * D_INNER / 128, NROWS / 256), 256, 0, stream>>>(
        x, W_in, xz, NROWS, 2 * D_INNER, MODEL_DIM);

    // 2) conv + silu
    {
        size_t total = (size_t)NROWS * D_INNER;
        conv_silu_kernel<<<(unsigned)((total + 255) / 256), 256, 0, stream>>>(
            xz, conv_w, conv_b, xconv);
    }

    // 3) ssm projection
    ssm_proj_kernel<<<NROWS, 256, 0, stream>>>(xconv, W_x, ssm);

    // 4) selective scan
    scan_kernel<<<dim3(D_INNER / 64, BATCH), 64, 0, stream>>>(
        xconv, ssm, dt_w, dt_b, A_log, Dp, yinner);

    // 5) gating (writes over xconv buffer)
    {
        size_t total = (size_t)NROWS * D_INNER;
        gate_kernel<<<(unsigned)((total + 255) / 256), 256, 0, stream>>>(
            yinner, xz, ygated);
    }

    // 6) out = ygated @ W_out  (M=4096, N=1024, K=2048)
    wmma_gemm_f32<<<dim3(MODEL_DIM / 128, NROWS / 256), 256, 0, stream>>>(
        ygated, W_out, out, NROWS, MODEL_DIM, D_INNER);
}